// GCLayer_59605556134259
// MI455X (gfx1250) — compile-verified
//
#include <hip/hip_runtime.h>

typedef __attribute__((ext_vector_type(2))) float v2f;
typedef __attribute__((ext_vector_type(8))) float v8f;

#define HIDDEN 128

// ---------------------------------------------------------------------------
// Zero the output accumulator (harness poisons d_out; atomics need zeros).
// Regular temporal stores: leaves the 51.2MB out array dirty in L2, which is
// exactly where the subsequent atomic RMWs want it.
// ---------------------------------------------------------------------------
__global__ __launch_bounds__(256) void zero_kernel(float* __restrict__ out, long long n4) {
  long long i = (long long)blockIdx.x * blockDim.x + threadIdx.x;
  if (i < n4) {
    ((float4*)out)[i] = make_float4(0.f, 0.f, 0.f, 0.f);
  }
}

// ---------------------------------------------------------------------------
// XW = X @ W using V_WMMA_F32_16X16X4_F32 (exact fp32 GEMM).
// One wave (32 lanes) computes a 16x16 output tile; 8 waves/block cover all
// 128 output columns of a 16-row stripe.
//
// A (16x4 f32): lanes 0-15 hold M=0..15, K={k0,k0+1}; lanes 16-31 same rows,
//   K={k0+2,k0+3}.
// B (4x16 f32): lanes 0-15 hold N=0..15 with K rows k0,k0+1; lanes 16-31 the
//   same columns with K rows k0+2,k0+3.
// C/D (16x16 f32): VGPR r holds M=r (lanes 0-15) / M=8+r (lanes 16-31),
//   N=lane%16.
// ---------------------------------------------------------------------------
__global__ __launch_bounds__(256) void gemm_xw(const float* __restrict__ X,
                                               const float* __restrict__ W,
                                               float* __restrict__ O, int N) {
  const int lane = threadIdx.x & 31;
  const int wave = threadIdx.x >> 5;  // 0..7 -> 16-col tile within 128 cols
  const int half = lane >> 4;         // 0: K pair {0,1}, 1: K pair {2,3}
  const int l16  = lane & 15;
  const int mtile = blockIdx.x;

  const int row = mtile * 16 + l16;   // A-matrix row for this lane
  const int col = wave * 16 + l16;    // B/C/D column for this lane
  const float* xrow = X + (size_t)row * HIDDEN;

  v8f acc = {};
#pragma unroll
  for (int k0 = 0; k0 < HIDDEN; k0 += 4) {
    const int ka = k0 + 2 * half;
    v2f a = *(const v2f*)(xrow + ka);          // X[row][ka..ka+1]
    v2f b;
    b.x = W[(size_t)ka * HIDDEN + col];
    b.y = W[(size_t)(ka + 1) * HIDDEN + col];
    // 8 args: (neg_a, A, neg_b, B, c_mod, C, reuse_a, reuse_b)
    acc = __builtin_amdgcn_wmma_f32_16x16x4_f32(false, a, false, b,
                                                (short)0, acc, false, false);
  }

#pragma unroll
  for (int r = 0; r < 8; ++r) {
    const int m = mtile * 16 + half * 8 + r;
    O[(size_t)m * HIDDEN + col] = acc[r];
  }
}

// ---------------------------------------------------------------------------
// For each edge e: out[dst[e]][:] += XW[src[e]][:]  (128 floats per edge).
//
// Each wave processes batches of 32 edges:
//   - lane l loads edge (eb+l)'s src/dst once (coalesced b64, non-temporal so
//     the streamed edge lists don't evict the L2-resident XW/out tables),
//   - the j-loop broadcasts indices via v_readlane (uniform j -> SGPR), so the
//     512B row gather is a single global_load_b128 with scalar base, and the
//     accumulation is 4 global_atomic_add_f32 with scalar base per lane,
//   - lanes cooperatively prefetch the row needed 6 edges ahead (each lane
//     covers 16B, 32 lanes span the full 512B row).
// ---------------------------------------------------------------------------
__global__ __launch_bounds__(256) void scatter_add(const float* __restrict__ XW,
                                                   const long long* __restrict__ src,
                                                   const long long* __restrict__ dst,
                                                   float* __restrict__ out,
                                                   long long E) {
  const int lane = threadIdx.x & 31;
  const long long wavesPerBlock = blockDim.x >> 5;
  const long long wid = (long long)blockIdx.x * wavesPerBlock + (threadIdx.x >> 5);
  const long long waveStride = (long long)gridDim.x * wavesPerBlock;

  for (long long eb = wid * 32; eb < E; eb += waveStride * 32) {
    // Per-lane edge indices for this batch of 32 edges.
    int myS = 0, myD = 0;
    const long long ee = eb + lane;
    if (ee < E) {
      myS = (int)__builtin_nontemporal_load(src + ee);
      myD = (int)__builtin_nontemporal_load(dst + ee);
    }
    const long long rem = E - eb;
    const int cnt = (rem < 32) ? (int)rem : 32;

    for (int j = 0; j < cnt; ++j) {
      // Prefetch the source row 6 edges ahead (full 512B across the wave).
      if (j + 6 < cnt) {
        const int spf = __builtin_amdgcn_readlane(myS, j + 6);
        __builtin_prefetch(XW + (size_t)spf * HIDDEN + lane * 4, 0, 0);
      }
      const int s = __builtin_amdgcn_readlane(myS, j);  // SGPR broadcast
      const int d = __builtin_amdgcn_readlane(myD, j);

      const float4 v = ((const float4*)(XW + (size_t)s * HIDDEN))[lane];
      float* o = out + (size_t)d * HIDDEN + lane * 4;
      unsafeAtomicAdd(o + 0, v.x);
      unsafeAtomicAdd(o + 1, v.y);
      unsafeAtomicAdd(o + 2, v.z);
      unsafeAtomicAdd(o + 3, v.w);
    }
  }
}

// ---------------------------------------------------------------------------
// Inputs (setup_inputs order):
//   0: hidden_states [100000,128] f32
//   1..4: edges_i/ii/iii/a [2,E] int64 (row 0 = src, row 1 = dst)
//   5..8: W_i/ii/iii/a [128,128] f32
// Output: [100000,128] f32.
// Workspace: N * 128 floats (single XW table, 51.2 MB, reused per edge set
// so that XW + out stay L2-resident during every scatter phase).
// ---------------------------------------------------------------------------
extern "C" void kernel_launch(void* const* d_in, const int* in_sizes, int n_in,
                              void* d_out, int out_size, void* d_ws, size_t ws_size,
                              hipStream_t stream) {
  const float* X = (const float*)d_in[0];
  const long long* edges[4] = {
      (const long long*)d_in[1], (const long long*)d_in[2],
      (const long long*)d_in[3], (const long long*)d_in[4]};
  const float* Ws[4] = {
      (const float*)d_in[5], (const float*)d_in[6],
      (const float*)d_in[7], (const float*)d_in[8]};
  float* out = (float*)d_out;
  float* xw  = (float*)d_ws;

  const int N = in_sizes[0] / HIDDEN;

  // 1) zero the accumulator
  const long long n4 = (long long)N * HIDDEN / 4;
  zero_kernel<<<(unsigned)((n4 + 255) / 256), 256, 0, stream>>>(out, n4);

  // 2) per edge type: fp32 WMMA GEMM into the shared table, then scatter-add.
  for (int t = 0; t < 4; ++t) {
    gemm_xw<<<(N + 15) / 16, 256, 0, stream>>>(X, Ws[t], xw, N);

    const long long E = in_sizes[1 + t] / 2;
    scatter_add<<<2048, 256, 0, stream>>>(xw, edges[t], edges[t] + E, out, E);
  }
}